// DarcyLoss_35407710388666
// MI455X (gfx1250) — compile-verified
//
#include <hip/hip_runtime.h>
#include <cstdint>

// DarcyLoss on a graph: scatter-mean over 4 edge groups + per-node combine.
// CDNA5 (gfx1250) paths used: GLOBAL_LOAD_ASYNC_TO_LDS_B128 + s_wait_asynccnt
// for streaming edge_attr tiles through LDS, global_prefetch_b8 for the
// edge_index stream, and no-return global_atomic_add_f32 for the scatter.

constexpr float DELTA_INV = 10.0f;   // 1/0.1
constexpr float F_SOURCE  = 1.0f;

constexpr int TPB  = 256;            // 8 wave32 per block
constexpr int EPT  = 2;              // edges per thread (16B edge_attr per lane)
constexpr int TILE = TPB * EPT;      // 512 edges per block-tile

// ---------------- Atomic lowering probe (defined FIRST so the disasm
// snippet shows exactly how atomicAdd(float*) lowers on gfx1250; never
// launched by kernel_launch) ------------------------------------------------

__global__ void darcy_atomic_probe(float* p, float v) {
    atomicAdd(p + threadIdx.x, v);
}

// ---------------- CDNA5 feature shims (builtin if present, else asm) -------

// Builtin signature (probed from clang diagnostic):
//   void __builtin_amdgcn_global_load_async_to_lds_b128(
//       int __attribute__((vector_size(16))) __device__ *src,   // AS(1)
//       int __attribute__((vector_size(16))) __local__  *dst,   // AS(3)
//       imm int offset, imm int cpol);
typedef int v4i_gnu __attribute__((vector_size(16)));
typedef __attribute__((address_space(1))) v4i_gnu* as1_v4i_p;
typedef __attribute__((address_space(3))) v4i_gnu* as3_v4i_p;

__device__ __forceinline__ void cdna5_async_load_b128_to_lds(const void* gsrc,
                                                             void* lds_dst) {
#if __has_builtin(__builtin_amdgcn_global_load_async_to_lds_b128)
    __builtin_amdgcn_global_load_async_to_lds_b128(
        (as1_v4i_p)(uintptr_t)gsrc,
        (as3_v4i_p)(uint32_t)(uintptr_t)lds_dst,
        0, 0);
#else
    asm volatile("global_load_async_to_lds_b128 %0, %1, off"
                 :
                 : "v"((uint32_t)(uintptr_t)lds_dst),
                   "v"((uint64_t)(uintptr_t)gsrc)
                 : "memory");
#endif
}

__device__ __forceinline__ void cdna5_wait_asynccnt0() {
#if __has_builtin(__builtin_amdgcn_s_wait_asynccnt)
    __builtin_amdgcn_s_wait_asynccnt(0);
#else
    asm volatile("s_wait_asynccnt 0" ::: "memory");
#endif
}

// ---------------- Pass 1: per-edge scatter accumulate ----------------------
// ws layout per node (12 floats, 48B, 16B aligned):
//   [0..2]  xp: sum_a, sum_d, cnt
//   [3..5]  xm: sum_a, sum_d, cnt
//   [6..8]  yp: sum_a, sum_d, cnt
//   [9..11] ym: sum_a, sum_d, cnt

__global__ __launch_bounds__(TPB) void darcy_edge_kernel(
    const float* __restrict__ u,
    const float* __restrict__ a,
    const float* __restrict__ edge_attr,   // (E,2) row-major
    const int*   __restrict__ edge_index,  // (2,E): [0,E)=src, [E,2E)=dst
    float*       __restrict__ ws,
    int E)
{
    __shared__ __align__(16) float lds_attr[TILE * 2];   // 4 KB / block

    const int t        = threadIdx.x;
    const int tileBase = blockIdx.x * TILE;
    const int* __restrict__ src_row = edge_index;
    const int* __restrict__ dst_row = edge_index + E;

    const bool fullTile = (tileBase + TILE) <= E;
    if (fullTile) {
        // Each lane stages its own 16B (2 edges x float2 attr) into LDS.
        cdna5_async_load_b128_to_lds(edge_attr + 2 * (size_t)(tileBase + EPT * t),
                                     &lds_attr[4 * t]);
        // Prefetch next tile's edge_index stream into L1 while we wait.
        const int nb = tileBase + TILE;
        if (nb + TILE <= E) {
            __builtin_prefetch(&src_row[nb + t], 0, 1);
            __builtin_prefetch(&dst_row[nb + t], 0, 1);
        }
        cdna5_wait_asynccnt0();   // each lane reads only what it staged
    }

#pragma unroll
    for (int k = 0; k < EPT; ++k) {
        const int e = tileBase + EPT * t + k;
        if (e < E) {
            float ax, ay;
            if (fullTile) {
                ax = lds_attr[4 * t + 2 * k + 0];
                ay = lds_attr[4 * t + 2 * k + 1];
            } else {
                ax = edge_attr[2 * (size_t)e + 0];
                ay = edge_attr[2 * (size_t)e + 1];
            }

            int   g;
            float denom;
            bool  valid = true;
            if      (ax > 0.0f) { g = 0; denom = ax; }
            else if (ax < 0.0f) { g = 1; denom = ax; }
            else if (ay > 0.0f) { g = 2; denom = ay; }
            else if (ay < 0.0f) { g = 3; denom = ay; }
            else                { g = 0; denom = 1.0f; valid = false; }

            if (valid) {
                const int s = src_row[e];
                const int d = dst_row[e];
                const float aval = a[s];          // L2-resident gather
                const float du   = u[d] - u[s];   // L2-resident gathers
                const float dv   = du / denom;

                float* base = ws + (size_t)d * 12u + (size_t)g * 3u;
                atomicAdd(base + 0, aval);   // -> global_atomic_add_f32 (no rtn)
                atomicAdd(base + 1, dv);
                atomicAdd(base + 2, 1.0f);
            }
        }
    }
}

// ---------------- Pass 2: per-node combine ---------------------------------

__global__ __launch_bounds__(256) void darcy_node_kernel(
    const float* __restrict__ ws,
    float*       __restrict__ out,
    int N)
{
    const int i = blockIdx.x * blockDim.x + threadIdx.x;
    if (i >= N) return;

    const float4* w = reinterpret_cast<const float4*>(ws + (size_t)i * 12u);
    const float4 w0 = w[0];   // xp.sa xp.sd xp.c | xm.sa
    const float4 w1 = w[1];   // xm.sd xm.c | yp.sa yp.sd
    const float4 w2 = w[2];   // yp.c | ym.sa ym.sd ym.c

    auto term = [](float sa, float sd, float c) -> float {
        if (c <= 0.0f) return 0.0f;          // masked mean of empty group -> 0
        const float inv = 1.0f / c;
        return (sa * inv) * (sd * inv);
    };

    const float acc = term(w0.x, w0.y, w0.z)    // +xp
                    - term(w0.w, w1.x, w1.y)    // -xm
                    + term(w1.z, w1.w, w2.x)    // +yp
                    - term(w2.y, w2.z, w2.w);   // -ym

    out[i] = acc * DELTA_INV + F_SOURCE;
}

// ---------------- Launch ----------------------------------------------------

extern "C" void kernel_launch(void* const* d_in, const int* in_sizes, int n_in,
                              void* d_out, int out_size, void* d_ws, size_t ws_size,
                              hipStream_t stream)
{
    const float* u         = (const float*)d_in[0];
    const float* a         = (const float*)d_in[1];
    const float* edge_attr = (const float*)d_in[2];
    const int*   edge_idx  = (const int*)d_in[3];
    float*       out       = (float*)d_out;
    float*       ws        = (float*)d_ws;

    const int N = in_sizes[0];        // nodes
    const int E = in_sizes[3] / 2;    // edges (edge_index is 2*E ints)

    const size_t wsBytes = (size_t)N * 12u * sizeof(float);  // 48 MB for 1M nodes
    (void)hipMemsetAsync(d_ws, 0, wsBytes, stream);          // graph-capturable

    const int numTiles = (E + TILE - 1) / TILE;
    darcy_edge_kernel<<<numTiles, TPB, 0, stream>>>(u, a, edge_attr, edge_idx, ws, E);

    darcy_node_kernel<<<(N + 255) / 256, 256, 0, stream>>>(ws, out, N);
}